// RBFLayer_keops_82377472737364
// MI455X (gfx1250) — compile-verified
//
#include <hip/hip_runtime.h>

// RBF mixture layer for MI455X (gfx1250, wave32).
//
// D_ij = (x_i - mu_m)^T (G_m G_m^T) (x_i - mu_m)
//      = [x_i (x) x_i , x_i] . [flatten(C_m), -2 C_m mu_m]  +  mu^T C mu
// out_i = sum_m (w_m e^{-c_m}) * exp(-quad_im)
//
// 142 GFLOP fp32 contraction on V_WMMA_F32_16X16X4_F32 (native fp32 WMMA keeps
// exp() numerically sane). All operands resident in 192MB L2.
// v3: 4 row-tiles x 4 m-tiles per wave -> 128 acc VGPRs (stays under the
// 256-VGPR window so the scheduler can multi-buffer B loads), each B fragment
// feeds 4 WMMAs (32 FLOP per L2 byte, 0.25 loads per wmma).

#define N_PTS   32768
#define M_COMP  2048
#define DIM     32
#define KTOT    1056              // 32*32 + 32
#define KC      132               // K processed in 8 chunks (LDS budget, 64 rows)
#define NCHUNK  8
#define FSTR    134               // LDS row stride: 134 mod 64 = 6 -> 16-row reads conflict-free
#define MSPLIT  8                 // grid.y split over M
#define WAVES   4                 // waves per block
#define P_TOT   (MSPLIT * WAVES)  // 32 partial buffers per output row
#define GP_FLOATS (528 * M_COMP * 2)  // 528 k-pairs * 2048 m * 2 floats

typedef float v2f __attribute__((ext_vector_type(2)));
typedef float v8f __attribute__((ext_vector_type(8)));

// ---------------------------------------------------------------------------
// Kernel 1: per-component precompute.
//   C_m = gamma gamma^T, b_m = C_m mu_m, c_m = b.mu, wexp_m = w_m * exp(-c_m)
//   Stored as GP[p][m][j] = G_m[2p+j]   (p = k/2)  -> coalesced b64 B-fragments
// ---------------------------------------------------------------------------
__global__ void rbf_prep(const float* __restrict__ gamma,
                         const float* __restrict__ means,
                         const float* __restrict__ wts,
                         float* __restrict__ gp,
                         float* __restrict__ wexp) {
  __shared__ float g[DIM * DIM];
  __shared__ float Cs[DIM * DIM];
  __shared__ float bs[DIM];
  const int m = blockIdx.x;
  const int tid = threadIdx.x;

  for (int idx = tid; idx < DIM * DIM; idx += blockDim.x)
    g[idx] = gamma[m * DIM * DIM + idx];
  __syncthreads();

  for (int idx = tid; idx < DIM * DIM; idx += blockDim.x) {
    const int d = idx >> 5, e = idx & 31;
    float s = 0.f;
    for (int k = 0; k < DIM; ++k) s = fmaf(g[d * DIM + k], g[e * DIM + k], s);
    Cs[idx] = s;
    gp[(size_t)(idx >> 1) * (2 * M_COMP) + m * 2 + (idx & 1)] = s;
  }
  __syncthreads();

  if (tid < DIM) {
    float b = 0.f;
    for (int e = 0; e < DIM; ++e) b = fmaf(Cs[tid * DIM + e], means[m * DIM + e], b);
    bs[tid] = b;
    const int k = DIM * DIM + tid;                       // features 1024..1055
    gp[(size_t)(k >> 1) * (2 * M_COMP) + m * 2 + (k & 1)] = -2.f * b;
  }
  __syncthreads();

  if (tid == 0) {
    float c = 0.f;
    for (int d = 0; d < DIM; ++d) c = fmaf(bs[d], means[m * DIM + d], c);
    wexp[m] = wts[m] * __expf(-c);
  }
}

// ---------------------------------------------------------------------------
// Kernel 2: main fused GEMM + exp + weighted reduce.
//   block: 128 threads = 4 waves; one 64-row feature tile in LDS shared by all.
//   wave w: 4 row-tiles x 4 m-tiles (64 m) -> 16 f32 acc tiles (128 VGPRs);
//   every B fragment (global_load_b64) is consumed by 4 WMMAs.
// ---------------------------------------------------------------------------
__global__ void __launch_bounds__(128, 4) rbf_main(const float* __restrict__ X,
                                                   const float* __restrict__ gp,
                                                   const float* __restrict__ wexp,
                                                   float* __restrict__ partial) {
  __shared__ float xs[64 * DIM];     // 64 input rows (8 KB)
  __shared__ float fsh[64 * FSTR];   // feature chunk tile, padded rows (~34 KB)

  const int tid  = threadIdx.x;
  const int w    = tid >> 5;
  const int lane = tid & 31;
  const int half = lane >> 4;        // 0: K=0,1 / rows 0-7   1: K=2,3 / rows 8-15
  const int l15  = lane & 15;
  const int i0   = blockIdx.x * 64;
  const int mchunk = blockIdx.y * (M_COMP / MSPLIT) + w * (M_COMP / MSPLIT / WAVES);

  for (int idx = tid; idx < 64 * DIM; idx += 128)
    xs[idx] = X[(size_t)i0 * DIM + idx];

  v8f acc[16];                        // acc[rt*4 + mt]
  const v8f vzero = {0.f, 0.f, 0.f, 0.f, 0.f, 0.f, 0.f, 0.f};
#pragma unroll
  for (int t = 0; t < 16; ++t) acc[t] = vzero;

  // per-lane constant B offset (float2 units): column l15, p += half
  const int laneoff = l15 + half * M_COMP;

  for (int c = 0; c < NCHUNK; ++c) {
    __syncthreads();                       // xs ready / previous fsh consumed
    // build features f_i[k] for k in [132c, 132c+132), rows i0..i0+63
    for (int r = 0; r < 64; ++r) {
      for (int kk = tid; kk < KC; kk += 128) {
        const int k = c * KC + kk;
        float v;
        if (k < DIM * DIM) v = xs[r * DIM + (k >> 5)] * xs[r * DIM + (k & 31)];
        else               v = xs[r * DIM + (k - DIM * DIM)];
        fsh[r * FSTR + kk] = v;
      }
    }
    __syncthreads();

    for (int s = 0; s < KC / 4; ++s) {     // 33 K-steps of 4
      // A frags: row tiles at rows rt*16 + l15, K = s*4 + 2*half + {0,1}
      v2f a[4];
#pragma unroll
      for (int rt = 0; rt < 4; ++rt)
        a[rt] = *(const v2f*)&fsh[(rt * 16 + l15) * FSTR + s * 4 + 2 * half];

      const float* gpb =
          gp + (size_t)((66 * c + 2 * s) * M_COMP + mchunk + laneoff) * 2;
#pragma unroll
      for (int mt = 0; mt < 4; ++mt) {
        const v2f b = *(const v2f*)(gpb + mt * 32);    // +16 m -> 32 floats
#pragma unroll
        for (int rt = 0; rt < 4; ++rt)
          acc[rt * 4 + mt] = __builtin_amdgcn_wmma_f32_16x16x4_f32(
              false, a[rt], false, b, (short)0, acc[rt * 4 + mt], false, false);
      }
    }
  }

  // Epilogue: lane holds column m = base + l15; VGPR j holds row j + 8*half.
  float row[4][8];
#pragma unroll
  for (int rt = 0; rt < 4; ++rt)
#pragma unroll
    for (int j = 0; j < 8; ++j) row[rt][j] = 0.f;

#pragma unroll
  for (int mt = 0; mt < 4; ++mt) {
    const float wv = wexp[mchunk + mt * 16 + l15];
#pragma unroll
    for (int rt = 0; rt < 4; ++rt)
#pragma unroll
      for (int j = 0; j < 8; ++j)
        row[rt][j] += wv * __expf(-acc[rt * 4 + mt][j]);
  }

  const int p = blockIdx.y * WAVES + w;
#pragma unroll
  for (int rt = 0; rt < 4; ++rt) {
#pragma unroll
    for (int j = 0; j < 8; ++j) {
      float v = row[rt][j];
      v += __shfl_xor(v, 1, 32);           // reduce over the 16 m-columns
      v += __shfl_xor(v, 2, 32);
      v += __shfl_xor(v, 4, 32);
      v += __shfl_xor(v, 8, 32);
      if (l15 == 0)
        partial[(size_t)p * N_PTS + i0 + rt * 16 + half * 8 + j] = v;
    }
  }
}

// ---------------------------------------------------------------------------
// Kernel 3: deterministic partial reduction (no float atomics).
// ---------------------------------------------------------------------------
__global__ void rbf_reduce(const float* __restrict__ partial,
                           float* __restrict__ out) {
  const int i = blockIdx.x * blockDim.x + threadIdx.x;
  if (i < N_PTS) {
    float s = 0.f;
    for (int pp = 0; pp < P_TOT; ++pp) s += partial[(size_t)pp * N_PTS + i];
    out[i] = s;
  }
}

extern "C" void kernel_launch(void* const* d_in, const int* in_sizes, int n_in,
                              void* d_out, int out_size, void* d_ws, size_t ws_size,
                              hipStream_t stream) {
  (void)in_sizes; (void)n_in; (void)out_size; (void)ws_size;
  const float* X     = (const float*)d_in[0];   // [N, 32]
  const float* gamma = (const float*)d_in[1];   // [M, 32, 32]
  const float* means = (const float*)d_in[2];   // [M, 32]
  const float* wts   = (const float*)d_in[3];   // [M]
  float* out = (float*)d_out;

  float* gp      = (float*)d_ws;                // 528*2048*2 floats (8.65 MB)
  float* wexp    = gp + GP_FLOATS;              // 2048 floats
  float* partial = wexp + M_COMP;               // 32 * 32768 floats (4 MB)

  rbf_prep<<<M_COMP, 256, 0, stream>>>(gamma, means, wts, gp, wexp);
  rbf_main<<<dim3(N_PTS / 64, MSPLIT), 128, 0, stream>>>(X, gp, wexp, partial);
  rbf_reduce<<<N_PTS / 256, 256, 0, stream>>>(partial, out);
}